// GS_40080634806827
// MI455X (gfx1250) — compile-verified
//
#include <hip/hip_runtime.h>
#include <math.h>

#define N_NODES 100000
#define N_EDGES 1600000
#define IN_C    256
#define HID_C   128
#define N_LAYERS 4
#define ALPHA   0.1f

typedef float v2f __attribute__((ext_vector_type(2)));
typedef float v8f __attribute__((ext_vector_type(8)));

__device__ __forceinline__ void fadd_agent(float* p, float v) {
    // relaxed, agent-scope fp32 atomic -> native global_atomic_add_f32 on gfx1250
    __hip_atomic_fetch_add(p, v, __ATOMIC_RELAXED, __HIP_MEMORY_SCOPE_AGENT);
}

// ---------------------------------------------------------------- degree prep
__global__ void kdeg_init(int* degi) {
    int v = blockIdx.x * 256 + threadIdx.x;
    if (v < N_NODES) degi[v] = 1;            // self loop
}

__global__ void kdeg_count(const int* __restrict__ dst, int* __restrict__ degi) {
    int e = blockIdx.x * 256 + threadIdx.x;
    if (e < N_EDGES) atomicAdd(&degi[dst[e]], 1);
}

__global__ void kdinv(const int* __restrict__ degi, float* __restrict__ dinv) {
    int v = blockIdx.x * 256 + threadIdx.x;
    if (v < N_NODES) dinv[v] = rsqrtf((float)degi[v]);
}

__global__ void kzero4(float4* __restrict__ p, int n4) {
    int i = blockIdx.x * 256 + threadIdx.x;
    if (i < n4) p[i] = make_float4(0.f, 0.f, 0.f, 0.f);
}

// ---------------------------------------------------------- projection GEMM
// x0 = x @ proj_w + proj_b ; [100000 x 256] @ [256 x 128]
// block: 256 thr = 8 waves; block tile M=16, wave tile N=16; K loop via WMMA f32 16x16x4
#define LDA_P 260   // 16-row LDS tile, padded stride (floats)
__global__ void kproj(const float* __restrict__ x, const float* __restrict__ w,
                      const float* __restrict__ bias, float* __restrict__ x0) {
    __shared__ float sa[16 * LDA_P];
    const int t   = threadIdx.x;
    const int m0  = blockIdx.x * 16;

    // stage A tile 16x256 into LDS (1024 float4, 4 per thread)
    const float4* xf4 = (const float4*)x;   // row = 64 float4
    #pragma unroll
    for (int c = 0; c < 4; ++c) {
        int idx = c * 256 + t;
        int row = idx >> 6, q = idx & 63;
        float4 v = xf4[(size_t)(m0 + row) * 64 + q];
        float* d = &sa[row * LDA_P + q * 4];
        d[0] = v.x; d[1] = v.y; d[2] = v.z; d[3] = v.w;
    }
    __syncthreads();

    const int wave = t >> 5, lane = t & 31;
    const int half = lane >> 4, col = lane & 15;
    const int n0   = wave * 16;
    const int mrow = col;                    // A: lanes 0-15 and 16-31 both cover M=0..15

    v8f acc = {};
    #pragma unroll 8
    for (int k0 = 0; k0 < IN_C; k0 += 4) {
        int ka = k0 + 2 * half;
        v2f a = *(const v2f*)&sa[mrow * LDA_P + ka];
        v2f b;
        b.x = w[(size_t)ka * HID_C + n0 + col];
        b.y = w[(size_t)(ka + 1) * HID_C + n0 + col];
        acc = __builtin_amdgcn_wmma_f32_16x16x4_f32(false, a, false, b,
                                                    (short)0, acc, false, false);
    }

    float bv = bias[n0 + col];
    #pragma unroll
    for (int i = 0; i < 8; ++i) {
        int row = i + 8 * half;
        x0[(size_t)(m0 + row) * HID_C + n0 + col] = acc[i] + bv;
    }
}

// ------------------------------------------------------------- edge scatter
// one wave32 per edge: agg[dst] += norm * h[src]   (lane = one float4)
__global__ void kscatter(const int* __restrict__ src, const int* __restrict__ dst,
                         const float* __restrict__ dinv, const float* __restrict__ hin,
                         float* __restrict__ agg) {
    int wid  = (blockIdx.x * 256 + threadIdx.x) >> 5;   // edge id (exact: 200000 blocks)
    int lane = threadIdx.x & 31;
    int s = src[wid], d = dst[wid];
    float nrm = dinv[s] * dinv[d];
    float4 v = ((const float4*)(hin + (size_t)s * HID_C))[lane];
    float* ap = agg + (size_t)d * HID_C + lane * 4;
    fadd_agent(ap + 0, nrm * v.x);
    fadd_agent(ap + 1, nrm * v.y);
    fadd_agent(ap + 2, nrm * v.z);
    fadd_agent(ap + 3, nrm * v.w);
}

// ------------------------------------------------- fused combine + layer GEMM
// support = (1-a)*(agg + dinv^2 * h) + a*x0      (self-loop folded in)
// out     = (1-beta)*support + beta*(support @ W)  [+ relu]
#define LDA_L 132
__global__ void klayer(const float* __restrict__ agg, const float* __restrict__ hin,
                       const float* __restrict__ x0, const float* __restrict__ dinv,
                       const float* __restrict__ w, float beta, int do_relu,
                       float* __restrict__ hout) {
    __shared__ float sup[16 * LDA_L];
    const int t  = threadIdx.x;
    const int m0 = blockIdx.x * 16;

    // build support tile 16x128 in LDS (512 float4, 2 per thread)
    const float4* af4 = (const float4*)agg;
    const float4* hf4 = (const float4*)hin;
    const float4* xf4 = (const float4*)x0;
    #pragma unroll
    for (int c = 0; c < 2; ++c) {
        int idx = c * 256 + t;
        int row = idx >> 5, q = idx & 31;
        float dv = dinv[m0 + row];
        float d2 = dv * dv;
        size_t base = (size_t)(m0 + row) * 32 + q;
        float4 a = af4[base], h = hf4[base], x = xf4[base];
        float* dp = &sup[row * LDA_L + q * 4];
        dp[0] = (1.f - ALPHA) * (a.x + d2 * h.x) + ALPHA * x.x;
        dp[1] = (1.f - ALPHA) * (a.y + d2 * h.y) + ALPHA * x.y;
        dp[2] = (1.f - ALPHA) * (a.z + d2 * h.z) + ALPHA * x.z;
        dp[3] = (1.f - ALPHA) * (a.w + d2 * h.w) + ALPHA * x.w;
    }
    __syncthreads();

    const int wave = t >> 5, lane = t & 31;
    const int half = lane >> 4, col = lane & 15;
    const int n0   = wave * 16;
    const int mrow = col;

    v8f acc = {};
    #pragma unroll 8
    for (int k0 = 0; k0 < HID_C; k0 += 4) {
        int ka = k0 + 2 * half;
        v2f a = *(const v2f*)&sup[mrow * LDA_L + ka];
        v2f b;
        b.x = w[(size_t)ka * HID_C + n0 + col];
        b.y = w[(size_t)(ka + 1) * HID_C + n0 + col];
        acc = __builtin_amdgcn_wmma_f32_16x16x4_f32(false, a, false, b,
                                                    (short)0, acc, false, false);
    }

    #pragma unroll
    for (int i = 0; i < 8; ++i) {
        int row = i + 8 * half;
        float s = sup[row * LDA_L + n0 + col];
        float v = (1.f - beta) * s + beta * acc[i];
        if (do_relu) v = fmaxf(v, 0.f);
        hout[(size_t)(m0 + row) * HID_C + n0 + col] = v;
    }
}

// ---------------------------------------------------------------- launcher
extern "C" void kernel_launch(void* const* d_in, const int* in_sizes, int n_in,
                              void* d_out, int out_size, void* d_ws, size_t ws_size,
                              hipStream_t stream) {
    const float* x      = (const float*)d_in[0];
    const int*   ei     = (const int*)d_in[1];          // [2, E]
    const float* proj_w = (const float*)d_in[2];
    const float* proj_b = (const float*)d_in[3];
    const float* conv_w = (const float*)d_in[4];        // [4,128,128]
    float* out = (float*)d_out;

    const int* e_src = ei;
    const int* e_dst = ei + N_EDGES;

    const size_t NH = (size_t)N_NODES * HID_C;
    char* ws = (char*)d_ws;
    int*   degi = (int*)ws;                                  ws += ((size_t)N_NODES * 4 + 255) & ~255ull;
    float* dinv = (float*)ws;                                ws += ((size_t)N_NODES * 4 + 255) & ~255ull;
    float* x0   = (float*)ws;                                ws += NH * 4;
    float* h    = (float*)ws;                                ws += NH * 4;
    float* agg  = (float*)ws;

    // degrees -> dinv
    kdeg_init <<<(N_NODES + 255) / 256, 256, 0, stream>>>(degi);
    kdeg_count<<<(N_EDGES + 255) / 256, 256, 0, stream>>>(e_dst, degi);
    kdinv     <<<(N_NODES + 255) / 256, 256, 0, stream>>>(degi, dinv);

    // x0 = x @ proj_w + proj_b
    kproj<<<N_NODES / 16, 256, 0, stream>>>(x, proj_w, proj_b, x0);

    static const float betas[N_LAYERS] = {0.40546510f, 0.22314355f, 0.15415068f, 0.11778303f};
    const int n4 = (int)(NH / 4);

    for (int i = 0; i < N_LAYERS; ++i) {
        const float* hin  = (i == 0) ? x0 : h;
        float*       hout = (i == N_LAYERS - 1) ? out : h;
        kzero4  <<<(n4 + 255) / 256, 256, 0, stream>>>((float4*)agg, n4);
        kscatter<<<N_EDGES / 8, 256, 0, stream>>>(e_src, e_dst, dinv, hin, agg);
        klayer  <<<N_NODES / 16, 256, 0, stream>>>(agg, hin, x0, dinv,
                                                   conv_w + (size_t)i * HID_C * HID_C,
                                                   betas[i], (i != N_LAYERS - 1) ? 1 : 0, hout);
    }
}